// MultilabelLinear_14955076124937
// MI455X (gfx1250) — compile-verified
//
#include <hip/hip_runtime.h>
#include <math.h>

typedef __attribute__((ext_vector_type(2))) float v2f;
typedef __attribute__((ext_vector_type(8))) float v8f;

#define BDIM   65536
#define DDIM   512
#define NHEAD  64
#define JDIM   128            // N*2 output columns
#define MTILE  16
#define LSTR   520            // padded LDS row stride (floats), even

// ---------------------------------------------------------------------------
// Kernel 1: per-head spectral norm (1 power iteration) + pack Wsn into a
// WMMA-B-matrix friendly layout:  P[p*256 + j*2 + q] = W[j/2][j&1][2p+q]/sigma
// ---------------------------------------------------------------------------
__global__ void sn_prep_kernel(const float* __restrict__ W,
                               const float* __restrict__ u,
                               float* __restrict__ P) {
    __shared__ float s_sigma[NHEAD];
    const int t = threadIdx.x;
    if (t < NHEAD) {
        const float* W0 = W + (size_t)(t * 2 + 0) * DDIM;
        const float* W1 = W + (size_t)(t * 2 + 1) * DDIM;
        const float u0 = u[t * 2 + 0];
        const float u1 = u[t * 2 + 1];
        // pass 1: ||v||, v[d] = W0[d]*u0 + W1[d]*u1
        float nv = 0.f;
        for (int d = 0; d < DDIM; ++d) {
            float vd = W0[d] * u0 + W1[d] * u1;
            nv += vd * vd;
        }
        float inv = 1.0f / fmaxf(sqrtf(nv), 1e-12f);
        // pass 2: w = W v_hat  (recompute v on the fly)
        float w0 = 0.f, w1 = 0.f;
        for (int d = 0; d < DDIM; ++d) {
            float vh = (W0[d] * u0 + W1[d] * u1) * inv;
            w0 += W0[d] * vh;
            w1 += W1[d] * vh;
        }
        float inu = 1.0f / fmaxf(sqrtf(w0 * w0 + w1 * w1), 1e-12f);
        // sigma = u1_hat . (W v_hat)
        s_sigma[t] = w0 * (w0 * inu) + w1 * (w1 * inu);
    }
    __syncthreads();
    // pack scaled transposed weights: 512*128 floats
    for (int e = t; e < DDIM * JDIM; e += 256) {
        int p = e >> 8;          // k-pair index 0..255
        int r = e & 255;
        int j = r >> 1;          // output column 0..127
        int q = r & 1;
        int n = j >> 1;
        int o = j & 1;
        int d = 2 * p + q;
        P[e] = W[(size_t)(n * 2 + o) * DDIM + d] / s_sigma[n];
    }
}

// ---------------------------------------------------------------------------
// Kernel 2: fused GEMM (WMMA f32 16x16x4) + bias + argmax + CE-loss partials.
// Block = 256 threads = 8 waves; block owns 16 rows (M), wave owns 16 cols (J).
// ---------------------------------------------------------------------------
__global__ void __launch_bounds__(256)
fused_gemm_kernel(const float* __restrict__ feature,
                  const int* __restrict__ target,
                  const float* __restrict__ P,
                  const float* __restrict__ bias,
                  float* __restrict__ out_pred,
                  float* __restrict__ out_arg,
                  float* __restrict__ partials) {
    __shared__ float ldsF[MTILE * LSTR];   // feature tile, padded
    __shared__ float ldsP[MTILE * JDIM];   // logit tile
    __shared__ float ldsR[32];             // loss partial scratch

    const int t    = threadIdx.x;
    const int m0   = blockIdx.x * MTILE;
    const int wave = t >> 5;               // 0..7 -> column tile
    const int lane = t & 31;
    const int half = lane >> 4;            // 0: lanes 0-15, 1: lanes 16-31
    const int lr   = lane & 15;
    const int n0   = wave * 16;
    const int colJ = n0 + lr;

    // ---- stage 16x512 feature rows into LDS (float4, coalesced) ----
    const float4* f4 = (const float4*)(feature + (size_t)m0 * DDIM);
    #pragma unroll
    for (int i = 0; i < 8; ++i) {
        int e  = t + 256 * i;              // float4 index, 0..2047
        int r  = e >> 7;                   // row 0..15 (128 float4 per row)
        int c4 = e & 127;
        float4 v = f4[r * 128 + c4];
        int base = r * LSTR + c4 * 4;
        ldsF[base + 0] = v.x;
        ldsF[base + 1] = v.y;
        ldsF[base + 2] = v.z;
        ldsF[base + 3] = v.w;
    }
    __syncthreads();

    // ---- K-loop: v_wmma_f32_16x16x4_f32, 128 steps over K=512 ----
    // A 16x4 lane layout: lanes 0-15 hold {K,K+1}, lanes 16-31 hold {K+2,K+3}
    // B 4x16 lane layout mirrors A with N striped across lanes.
    v8f acc = {};
    const float* arow = ldsF + lr * LSTR;
    #pragma unroll 4
    for (int k = 0; k < DDIM; k += 4) {
        int kk = k + 2 * half;
        v2f a = *(const v2f*)(arow + kk);                       // ds_load_b64
        v2f b = *(const v2f*)(P + (kk >> 1) * 256 + colJ * 2);  // global b64
        acc = __builtin_amdgcn_wmma_f32_16x16x4_f32(
            /*neg_a=*/false, a, /*neg_b=*/false, b,
            /*c_mod=*/(short)0, acc, /*reuse_a=*/false, /*reuse_b=*/false);
    }

    // ---- epilogue: bias add, write pred, mirror tile to LDS ----
    const float bj = bias[colJ];
    #pragma unroll
    for (int r = 0; r < 8; ++r) {
        int row = half * 8 + r;            // C/D layout: VGPR r -> M = half*8+r
        float val = acc[r] + bj;
        out_pred[(size_t)(m0 + row) * JDIM + colJ] = val;
        ldsP[row * JDIM + colJ] = val;
    }
    __syncthreads();

    // ---- argmax over class pair: output[b][n] = pred[...,1] > pred[...,0] ----
    #pragma unroll
    for (int i = 0; i < 4; ++i) {
        int e   = t + 256 * i;             // 0..1023 = 16 rows * 64 heads
        int row = e >> 6;
        int n   = e & 63;
        float p0 = ldsP[row * JDIM + 2 * n + 0];
        float p1 = ldsP[row * JDIM + 2 * n + 1];
        out_arg[(size_t)(m0 + row) * NHEAD + n] = (p1 > p0) ? 1.0f : 0.0f;
    }

    // ---- CE loss with log_softmax over the head axis (axis=1 quirk) ----
    // per (row, o): contribution = cnt*logsumexp_n(pred[:,o]) - sum_{target==o} pred
    if (t < 32) {
        int row = t >> 1;
        int o   = t & 1;
        int brow = m0 + row;
        const float* pr = ldsP + row * JDIM + o;
        float mx = -INFINITY;
        for (int n = 0; n < NHEAD; ++n) mx = fmaxf(mx, pr[2 * n]);
        float se = 0.f, tsum = 0.f;
        int cnt = 0;
        for (int n = 0; n < NHEAD; ++n) {
            float p = pr[2 * n];
            se += __expf(p - mx);
            if (target[(size_t)brow * NHEAD + n] == o) { tsum += p; ++cnt; }
        }
        float lse = mx + __logf(se);
        ldsR[t] = (float)cnt * lse - tsum;
    }
    __syncthreads();
    if (t == 0) {
        float s = 0.f;
        #pragma unroll
        for (int i = 0; i < 32; ++i) s += ldsR[i];
        partials[blockIdx.x] = s;
    }
}

// ---------------------------------------------------------------------------
// Kernel 3: reduce per-block loss partials -> scalar loss
// ---------------------------------------------------------------------------
__global__ void loss_reduce_kernel(const float* __restrict__ partials,
                                   float* __restrict__ out_loss) {
    __shared__ float sh[256];
    float s = 0.f;
    for (int i = threadIdx.x; i < BDIM / MTILE; i += 256) s += partials[i];
    sh[threadIdx.x] = s;
    __syncthreads();
    for (int st = 128; st > 0; st >>= 1) {
        if (threadIdx.x < st) sh[threadIdx.x] += sh[threadIdx.x + st];
        __syncthreads();
    }
    if (threadIdx.x == 0) out_loss[0] = sh[0] / (float)(BDIM * 2);
}

extern "C" void kernel_launch(void* const* d_in, const int* in_sizes, int n_in,
                              void* d_out, int out_size, void* d_ws, size_t ws_size,
                              hipStream_t stream) {
    const float* feature = (const float*)d_in[0];  // [B, D]
    const int*   target  = (const int*)d_in[1];    // [B, N]
    const float* W       = (const float*)d_in[2];  // [N, 2, D]
    const float* bias    = (const float*)d_in[3];  // [N, 2]
    const float* u       = (const float*)d_in[4];  // [N, 2]

    float* out      = (float*)d_out;
    float* out_pred = out;                                   // B*128
    float* out_arg  = out + (size_t)BDIM * JDIM;             // B*64
    float* out_loss = out + (size_t)BDIM * JDIM + (size_t)BDIM * NHEAD;

    float* P        = (float*)d_ws;                          // 512*128 floats
    float* partials = P + DDIM * JDIM;                       // 4096 floats

    sn_prep_kernel<<<1, 256, 0, stream>>>(W, u, P);
    fused_gemm_kernel<<<BDIM / MTILE, 256, 0, stream>>>(
        feature, target, P, bias, out_pred, out_arg, partials);
    loss_reduce_kernel<<<1, 256, 0, stream>>>(partials, out_loss);
}